// MambaEncoderBlock_64768106824321
// MI455X (gfx1250) — compile-verified
//
#include <hip/hip_runtime.h>
#include <hip/hip_bf16.h>
#include <cstdint>

typedef float v2f __attribute__((ext_vector_type(2)));
typedef float v4f __attribute__((ext_vector_type(4)));
typedef float v8f __attribute__((ext_vector_type(8)));

#define D_MODEL 128
#define D_INNER 256
#define D_STATE 16
#define DT_RANK 8
#define B_SZ    4
#define L_SEQ   4096
#define NCHUNK  32
#define LCHUNK  128

// ---------------------------------------------------------------------------
// CDNA5 async global->LDS copy (ASYNCcnt) with portable fallback.
// Builtin signature (from hipcc diagnostic): first param is
//   'int __vector_size(16) __device__(AS1) *'  -- typed int4 global pointer.
// ---------------------------------------------------------------------------
#if __has_builtin(__builtin_amdgcn_global_load_async_to_lds_b128)
#define HAVE_ASYNC_B128 1
#else
#define HAVE_ASYNC_B128 0
#endif

typedef int v4i_b __attribute__((vector_size(16)));
typedef __attribute__((address_space(1))) v4i_b* as1_v4i;
typedef __attribute__((address_space(3))) v4i_b* as3_v4i;

__device__ __forceinline__ void cp16(float* lds_dst, const float* gsrc)
{
#if HAVE_ASYNC_B128
    __builtin_amdgcn_global_load_async_to_lds_b128(
        (as1_v4i)(uintptr_t)gsrc,
        (as3_v4i)(unsigned)(uintptr_t)lds_dst,
        0, 0);
#else
    *(v4f*)lds_dst = *(const v4f*)gsrc;
#endif
}

__device__ __forceinline__ void cp_wait()
{
#if HAVE_ASYNC_B128
#if __has_builtin(__builtin_amdgcn_s_wait_asynccnt)
    __builtin_amdgcn_s_wait_asynccnt(0);
#else
    asm volatile("s_wait_asynccnt 0x0" ::: "memory");
#endif
#endif
}

// ---------------------------------------------------------------------------
// 1) conv3x3 stride2 + bias + BN(eval) + exact GELU.
//    Writes y in NCHW (residual) and u in [B, L, C] (GEMM-ready layout).
// ---------------------------------------------------------------------------
__global__ __launch_bounds__(256) void k_conv_bn_gelu(
    const float* __restrict__ x, const float* __restrict__ cw,
    const float* __restrict__ cb, const float* __restrict__ g,
    const float* __restrict__ bb, const float* __restrict__ mean,
    const float* __restrict__ var, float* __restrict__ yo,
    float* __restrict__ uo)
{
    __shared__ float wsh[16 * 576];                      // 16 oc x (64 ic * 9)
    const int b = blockIdx.z, ocg = blockIdx.y, tile = blockIdx.x;
    const int oy0 = (tile >> 2) * 16, ox0 = (tile & 3) * 16;
    const int tid = threadIdx.x;
    for (int i = tid; i < 16 * 576; i += 256) wsh[i] = cw[ocg * 16 * 576 + i];
    __syncthreads();

    const int oy = oy0 + (tid >> 4), ox = ox0 + (tid & 15);
    const int iy0 = oy * 2 - 1, ix0 = ox * 2 - 1;
    const float* xb = x + (long)b * 64 * 128 * 128;

    float acc[16];
#pragma unroll
    for (int oc = 0; oc < 16; ++oc) acc[oc] = 0.f;

    for (int ic = 0; ic < 64; ++ic) {
        float p[9];
#pragma unroll
        for (int ky = 0; ky < 3; ++ky)
#pragma unroll
            for (int kx = 0; kx < 3; ++kx) {
                const int iy = iy0 + ky, ix = ix0 + kx;
                const bool ok = (iy >= 0) & (iy < 128) & (ix >= 0) & (ix < 128);
                // branchless: clamp address, select value
                const int iyc = iy < 0 ? 0 : (iy > 127 ? 127 : iy);
                const int ixc = ix < 0 ? 0 : (ix > 127 ? 127 : ix);
                const float v = xb[(long)ic * 16384 + iyc * 128 + ixc];
                p[ky * 3 + kx] = ok ? v : 0.f;
            }
#pragma unroll
        for (int oc = 0; oc < 16; ++oc) {
            const float* wp = &wsh[oc * 576 + ic * 9];
            float s = acc[oc];
#pragma unroll
            for (int j = 0; j < 9; ++j) s = fmaf(p[j], wp[j], s);
            acc[oc] = s;
        }
    }
    const int l = oy * 64 + ox;
#pragma unroll
    for (int oc = 0; oc < 16; ++oc) {
        const int c = ocg * 16 + oc;
        float v = acc[oc] + cb[c];
        v = (v - mean[c]) * (g[c] / sqrtf(var[c] + 1e-5f)) + bb[c];
        v = 0.5f * v * (1.f + erff(v * 0.70710678118654752f));   // exact GELU
        yo[((long)(b * 128 + c)) * 4096 + l] = v;
        uo[((long)(b * 4096 + l)) * 128 + c] = v;
    }
}

// ---------------------------------------------------------------------------
// 2) fp32 WMMA GEMM:  C[M,N] = A[M,K] * W[N,K]^T   (+ optional softplus+bias)
//    block tile 128x64, 8 waves of 32x32, K-chunks of 16.
//    ASYNC path: double-buffered global_load_async_to_lds_b128 staging.
//    Both LDS tiles are [row][k] with stride KPAD=20 (16B-aligned rows,
//    conflict-free b64 fragment reads: 20*row mod 64 distinct over 16 rows).
// ---------------------------------------------------------------------------
#define MT 128
#define NT 64
#define KC 16
#define KPAD 20

template <int EPI, bool ASYNC>
__global__ __launch_bounds__(256) void k_gemm_wmma_f32(
    const float* __restrict__ A, int lda, const float* __restrict__ W,
    const float* __restrict__ bias, float* __restrict__ C, int ldc,
    int M, int N, int K)
{
    __shared__ float As[2][MT * KPAD];
    __shared__ float Bs[2][NT * KPAD];
    const int tid = threadIdx.x;
    const int m0 = blockIdx.y * MT;
    const int n0 = blockIdx.x * NT;
    const int wave = tid >> 5, lane = tid & 31;
    const int wm = (wave & 3) * 32, wn = (wave >> 2) * 32;
    const int lhalf = lane >> 4, l15 = lane & 15;

    v8f acc[2][2] = {};

    auto stage_async = [&](int buf, int k0) {
#pragma unroll
        for (int p = 0; p < 2; ++p) {                // A: 2 x 16B per thread
            const int ci = tid + p * 256;
            const int r = ci >> 2, q = ci & 3;
            cp16(&As[buf][r * KPAD + q * 4],
                 A + (long)(m0 + r) * lda + k0 + q * 4);
        }
        {                                            // B: 1 x 16B per thread
            const int nn = tid & 63, q = tid >> 6;
            const int ng = (n0 + nn < N) ? (n0 + nn) : (N - 1);  // clamp; junk
            cp16(&Bs[buf][nn * KPAD + q * 4],        // cols >= N die at store
                 W + (long)ng * K + k0 + q * 4);
        }
    };

    auto compute = [&](int buf) {
#pragma unroll
        for (int ks = 0; ks < KC; ks += 4) {
            v2f a[2], bf[2];
#pragma unroll
            for (int mi = 0; mi < 2; ++mi) {
                const float* p =
                    &As[buf][(wm + mi * 16 + l15) * KPAD + ks + lhalf * 2];
                a[mi].x = p[0];
                a[mi].y = p[1];
            }
#pragma unroll
            for (int ni = 0; ni < 2; ++ni) {
                const float* p =
                    &Bs[buf][(wn + ni * 16 + l15) * KPAD + ks + lhalf * 2];
                bf[ni].x = p[0];
                bf[ni].y = p[1];
            }
#pragma unroll
            for (int mi = 0; mi < 2; ++mi)
#pragma unroll
                for (int ni = 0; ni < 2; ++ni)
                    acc[mi][ni] = __builtin_amdgcn_wmma_f32_16x16x4_f32(
                        false, a[mi], false, bf[ni], (short)0, acc[mi][ni],
                        false, false);
        }
    };

    if (ASYNC) {
        // requires: K % KC == 0, A rows 16B-aligned (lda % 4 == 0)
        stage_async(0, 0);
        cp_wait();
        __syncthreads();
        const int nk = K / KC;
        for (int ki = 0; ki < nk; ++ki) {
            if (ki + 1 < nk) stage_async((ki + 1) & 1, (ki + 1) * KC);
            compute(ki & 1);
            cp_wait();
            __syncthreads();
        }
    } else {
        // guarded scalar staging (branchless clamp + select, no exec branches)
        for (int k0 = 0; k0 < K; k0 += KC) {
            {
                const int r = tid >> 1, kp = (tid & 1) * 8;
                const float* ap = A + (long)(m0 + r) * lda;
#pragma unroll
                for (int i = 0; i < 8; ++i) {
                    const int kk = k0 + kp + i;
                    const int kcl = kk < K ? kk : K - 1;
                    const float v = ap[kcl];
                    As[0][r * KPAD + kp + i] = kk < K ? v : 0.f;
                }
            }
            {
                const int nn = tid & 63, q = tid >> 6;
                const int ng = (n0 + nn < N) ? (n0 + nn) : (N - 1);
                const float* wp = W + (long)ng * K;
#pragma unroll
                for (int i = 0; i < 4; ++i) {
                    const int kk = k0 + q * 4 + i;
                    const int kcl = kk < K ? kk : K - 1;
                    const float v = wp[kcl];
                    Bs[0][nn * KPAD + q * 4 + i] = kk < K ? v : 0.f;
                }
            }
            __syncthreads();
            compute(0);
            __syncthreads();
        }
    }

#pragma unroll
    for (int mi = 0; mi < 2; ++mi)
#pragma unroll
        for (int ni = 0; ni < 2; ++ni) {
            const int colg = n0 + wn + ni * 16 + l15;
            if (colg < N) {
#pragma unroll
                for (int r = 0; r < 8; ++r) {
                    const int rowg = m0 + wm + mi * 16 + r + lhalf * 8;
                    float v = acc[mi][ni][r];
                    if (EPI == 1) {
                        v += bias[colg];
                        v = fmaxf(v, 0.f) + log1pf(expf(-fabsf(v)));  // softplus
                    }
                    C[(long)rowg * ldc + colg] = v;
                }
            }
        }
}

// ---------------------------------------------------------------------------
// 3) depthwise causal conv1d over L + SiLU.  block = 256 channels, grid = B*L
// ---------------------------------------------------------------------------
__global__ __launch_bounds__(256) void k_conv1d_silu(
    const float* __restrict__ xz, const float* __restrict__ w,
    const float* __restrict__ bias, float* __restrict__ xin)
{
    const int bl = blockIdx.x;
    const int d = threadIdx.x;
    const int b = bl >> 12, l = bl & 4095;
    float acc = bias[d];
#pragma unroll
    for (int k = 0; k < 4; ++k) {
        const int t = l - 3 + k;
        const int tc = t < 0 ? 0 : t;                     // branchless tap
        const float v = xz[((long)(b * 4096 + tc)) * 512 + d];
        acc = fmaf(t >= 0 ? w[d * 4 + k] : 0.f, v, acc);
    }
    xin[(long)bl * 256 + d] = acc / (1.f + __expf(-acc));
}

// ---------------------------------------------------------------------------
// 4a) chunked scan pass A: per (b,chunk,d,n) chunk transfer (P, S):
//     h_end = P * h_begin + S  for  h_t = a_t h + b_t ,  a = exp(delta*A)
// ---------------------------------------------------------------------------
__global__ __launch_bounds__(256) void k_scan_stats(
    const float* __restrict__ delta, const float* __restrict__ xin,
    const float* __restrict__ xdbl, const float* __restrict__ A_log,
    float* __restrict__ P, float* __restrict__ S)
{
    const int idx = blockIdx.x;                 // (b, chunk, dgrp)
    const int dgrp = idx & 15, c = (idx >> 4) & 31, b = idx >> 9;
    const int tid = threadIdx.x;
    const int n = tid & 15, d = dgrp * 16 + (tid >> 4);
    const float Aa = -__expf(A_log[d * 16 + n]);
    float Pp = 1.f, Ss = 0.f;
    const long base = (long)(b * 4096 + c * LCHUNK);
    for (int t = 0; t < LCHUNK; ++t) {
        const long bt = base + t;
        const float dl = delta[bt * 256 + d];
        const float xv = xin[bt * 256 + d];
        const float Bm = xdbl[bt * 40 + 8 + n];
        const float a = __expf(dl * Aa);
        Ss = fmaf(a, Ss, dl * Bm * xv);
        Pp *= a;
    }
    const long o = ((long)((b * 32 + c) * 256 + d)) * 16 + n;
    P[o] = Pp;
    S[o] = Ss;
}

// 4b) prefix over the 32 chunks -> initial state per chunk
__global__ __launch_bounds__(256) void k_scan_prefix(
    const float* __restrict__ P, const float* __restrict__ S,
    float* __restrict__ hinit)
{
    const int id = blockIdx.x * 256 + threadIdx.x;     // (b,d,n)
    const int n = id & 15, d = (id >> 4) & 255, b = id >> 12;
    float h = 0.f;
    for (int c = 0; c < NCHUNK; ++c) {
        const long o = ((long)((b * 32 + c) * 256 + d)) * 16 + n;
        hinit[o] = h;
        h = fmaf(P[o], h, S[o]);
    }
}

// 4c) re-apply within chunk with correct h0; fused epilogue:
//     y = (sum_n h*C) + x*Dp ;  y *= silu(z) ;  written in place over delta
__global__ __launch_bounds__(256) void k_scan_apply(
    const float* __restrict__ xdbl, const float* __restrict__ xin,
    const float* __restrict__ xz, const float* __restrict__ A_log,
    const float* __restrict__ Dp, const float* __restrict__ hinit,
    float* __restrict__ dy)
{
    const int c = blockIdx.x & 31, b = blockIdx.x >> 5;
    const int d = threadIdx.x;
    float h[16], Aa[16];
#pragma unroll
    for (int n = 0; n < 16; ++n) {
        h[n] = hinit[((long)((b * 32 + c) * 256 + d)) * 16 + n];
        Aa[n] = -__expf(A_log[d * 16 + n]);
    }
    const float dp = Dp[d];
    const long base = (long)(b * 4096 + c * LCHUNK);
    for (int t = 0; t < LCHUNK; ++t) {
        const long bt = base + t;
        const float dl = dy[bt * 256 + d];
        const float xv = xin[bt * 256 + d];
        const float zv = xz[bt * 512 + 256 + d];
        float y = 0.f;
#pragma unroll
        for (int n = 0; n < 16; ++n) {
            const float a = __expf(dl * Aa[n]);
            h[n] = fmaf(a, h[n], dl * xdbl[bt * 40 + 8 + n] * xv);
            y = fmaf(h[n], xdbl[bt * 40 + 24 + n], y);
        }
        y = fmaf(xv, dp, y);
        const float sz = zv / (1.f + __expf(-zv));
        dy[bt * 256 + d] = y * sz;
    }
}

// ---------------------------------------------------------------------------
// 5) final: d_out[b,c,l] = y[b,c,l] + outbuf[b,l,c]  (LDS 32x32 transpose)
// ---------------------------------------------------------------------------
__global__ __launch_bounds__(256) void k_transpose_add(
    const float* __restrict__ y, const float* __restrict__ ob,
    float* __restrict__ out)
{
    __shared__ float t[32][33];
    const int bidx = blockIdx.x;
    const int lt = bidx & 127, ct = (bidx >> 7) & 3, b = bidx >> 9;
    const int l0 = lt * 32, c0 = ct * 32;
    const int tid = threadIdx.x;
    const int cc = tid & 31, r0 = tid >> 5;
#pragma unroll
    for (int p = 0; p < 4; ++p) {
        const int r = r0 + p * 8;
        t[r][cc] = ob[((long)(b * 4096 + l0 + r)) * 128 + c0 + cc];
    }
    __syncthreads();
    const int lc = tid & 31, cr0 = tid >> 5;
#pragma unroll
    for (int p = 0; p < 4; ++p) {
        const int cr = cr0 + p * 8;
        const long idx = ((long)(b * 128 + c0 + cr)) * 4096 + l0 + lc;
        out[idx] = y[idx] + t[lc][cr];
    }
}

// ---------------------------------------------------------------------------
extern "C" void kernel_launch(void* const* d_in, const int* in_sizes, int n_in,
                              void* d_out, int out_size, void* d_ws,
                              size_t ws_size, hipStream_t stream)
{
    const float* x        = (const float*)d_in[0];
    const float* conv_w   = (const float*)d_in[1];
    const float* conv_b   = (const float*)d_in[2];
    const float* bn_g     = (const float*)d_in[3];
    const float* bn_b     = (const float*)d_in[4];
    const float* bn_mean  = (const float*)d_in[5];
    const float* bn_var   = (const float*)d_in[6];
    const float* in_w     = (const float*)d_in[7];
    const float* conv1d_w = (const float*)d_in[8];
    const float* conv1d_b = (const float*)d_in[9];
    const float* xproj_w  = (const float*)d_in[10];
    const float* dtproj_w = (const float*)d_in[11];
    const float* dtproj_b = (const float*)d_in[12];
    const float* A_log    = (const float*)d_in[13];
    const float* Dp       = (const float*)d_in[14];
    const float* out_w    = (const float*)d_in[15];
    float* out = (float*)d_out;

    float* ws = (float*)d_ws;
    const long BL = (long)B_SZ * L_SEQ;           // 16384
    float* y_ws   = ws;                            // [B,128,64,64]      2.10M
    float* u_ws   = y_ws + BL * 128;               // [B,L,128] /outbuf  2.10M
    float* xz_ws  = u_ws + BL * 128;               // [B,L,512]          8.39M
    float* xin_ws = xz_ws + BL * 512;              // [B,L,256]          4.19M
    float* xdb_ws = xin_ws + BL * 256;             // [B,L,40]           0.66M
    float* dlt_ws = xdb_ws + BL * 40;              // [B,L,256] delta->y 4.19M
    float* P_ws   = dlt_ws + BL * 256;             // [B,32,256,16]      0.52M
    float* S_ws   = P_ws + (long)B_SZ * 32 * 256 * 16;
    float* h_ws   = S_ws + (long)B_SZ * 32 * 256 * 16;

    const int M = (int)BL;

    // 1) conv stem (+ transposed copy for GEMM)
    k_conv_bn_gelu<<<dim3(16, 8, B_SZ), 256, 0, stream>>>(
        x, conv_w, conv_b, bn_g, bn_b, bn_mean, bn_var, y_ws, u_ws);

    // 2) in_proj: xz[M,512] = u[M,128] * in_w[512,128]^T     (async path)
    k_gemm_wmma_f32<0, true><<<dim3(512 / NT, M / MT), 256, 0, stream>>>(
        u_ws, 128, in_w, nullptr, xz_ws, 512, M, 512, 128);

    // 3) depthwise causal conv1d + SiLU on x half of xz
    k_conv1d_silu<<<dim3(M), 256, 0, stream>>>(xz_ws, conv1d_w, conv1d_b,
                                               xin_ws);

    // 4) x_proj: x_dbl[M,40] = x_in[M,256] * xproj_w[40,256]^T (async path)
    k_gemm_wmma_f32<0, true><<<dim3(1, M / MT), 256, 0, stream>>>(
        xin_ws, 256, xproj_w, nullptr, xdb_ws, 40, M, 40, 256);

    // 5) dt_proj + softplus: delta[M,256] = softplus(dt[M,8]*W^T + b)
    //    K=8, lda=40 -> guarded scalar path (single K-chunk)
    k_gemm_wmma_f32<1, false><<<dim3(256 / NT, M / MT), 256, 0, stream>>>(
        xdb_ws, 40, dtproj_w, dtproj_b, dlt_ws, 256, M, 256, 8);

    // 6) chunked selective scan (A: stats, B: prefix, C: apply + epilogue)
    k_scan_stats<<<dim3(B_SZ * NCHUNK * 16), 256, 0, stream>>>(
        dlt_ws, xin_ws, xdb_ws, A_log, P_ws, S_ws);
    k_scan_prefix<<<dim3(64), 256, 0, stream>>>(P_ws, S_ws, h_ws);
    k_scan_apply<<<dim3(B_SZ * NCHUNK), 256, 0, stream>>>(
        xdb_ws, xin_ws, xz_ws, A_log, Dp, h_ws, dlt_ws);

    // 7) out_proj: outbuf[M,128] = ypost[M,256] * out_w[128,256]^T (async)
    k_gemm_wmma_f32<0, true><<<dim3(128 / NT, M / MT), 256, 0, stream>>>(
        dlt_ws, 256, out_w, nullptr, u_ws, 128, M, 128, 256);

    // 8) residual add with transpose back to NCHW
    k_transpose_add<<<dim3(2048), 256, 0, stream>>>(y_ws, u_ws, out);
}